// MedicalTransformer_85873576116991
// MI455X (gfx1250) — compile-verified
//
#include <hip/hip_runtime.h>

// ---------------- problem constants ----------------
constexpr int kN  = 4, kC = 64, kH = 128, kW = 128, kG = 8;
constexpr int kB  = kN * kW;     // 512 axial batches
constexpr int kC2 = 2 * kC;      // 128 qkv channels
constexpr float kEPS  = 1e-5f;
constexpr float kF_QR = 0.1f, kF_KR = 0.1f, kF_SVE = 0.1f, kF_SV = 1.0f;
constexpr float kLOG2E = 1.44269504088896341f;

// ---------------- workspace layout (floats) ----------------
constexpr size_t kQKVElems = (size_t)kB * kC2 * kH;      // 8,388,608
constexpr size_t OFF_QKV  = 0;                           // raw qkv (pre-BN)
constexpr size_t OFF_OUTR = OFF_QKV  + kQKVElems;        // raw out (pre-BN)
constexpr size_t OFF_SQKV = OFF_OUTR + kQKVElems;        // qkv stats: sum[128], sq[128]
constexpr size_t OFF_SSIM = OFF_SQKV + 2 * kC2;          // sim stats: sum[24], sq[24]
constexpr size_t OFF_SOUT = OFF_SSIM + 48;               // out stats: sum[128], sq[128]
constexpr size_t OFF_CQKV = OFF_SOUT + 2 * kC2;          // coef: scale[128], shift[128]
constexpr size_t OFF_CSIM = OFF_CQKV + 2 * kC2;          // coef: scale[24], shift[24]
constexpr size_t OFF_COUT = OFF_CSIM + 48;               // coef: scale[128], shift[128]
// total ~16.78M floats = ~67.1 MB of d_ws

typedef float v2f __attribute__((ext_vector_type(2)));
typedef float v8f __attribute__((ext_vector_type(8)));
typedef unsigned int u32x4 __attribute__((ext_vector_type(4)));
typedef int i32x4 __attribute__((ext_vector_type(4)));
typedef int i32x8 __attribute__((ext_vector_type(8)));

// LDS byte offset of a __shared__ pointer: flat LDS-aperture addresses carry the
// LDS offset in their low 32 bits (ISA 10.2 aperture mapping).
__device__ __forceinline__ unsigned int lds_off(const void* p) {
    return (unsigned int)(unsigned long long)(uintptr_t)p;
}

// ---------------- TDM: 2D tile DMA global -> LDS ----------------
// Loads tile_y rows of tile_x contiguous elements (4B each), row stride
// stride_elems, packed contiguously into LDS at lds_byte_off.
// D# per CDNA5 ISA ch.8: group0 = {count|flags, lds_addr, global_addr, type=2},
// group1 = {data_size, tensor dims (set huge: no OOB clipping), tile dims, stride}.
__device__ __forceinline__ void tdm_load_2d(const void* gptr, unsigned int lds_byte_off,
                                            unsigned int tile_x, unsigned int tile_y,
                                            unsigned int stride_elems) {
    const unsigned long long ga = (unsigned long long)(uintptr_t)gptr;
    u32x4 g0;
    g0.x = 1u;                                        // count=1, user descriptor
    g0.y = lds_byte_off;                              // bits 63:32 lds_addr
    g0.z = (unsigned int)ga;                          // global_addr[31:0]
    g0.w = (unsigned int)((ga >> 32) & 0x01FFFFFFu)   // global_addr[56:32]
         | (2u << 30);                                // type=2 ("image")
    i32x8 g1;
    g1[0] = (int)(2u << 16);                          // data_size=2 (4B), mask=0
    g1[1] = (int)(0xFFFFu << 16);                     // tensor_dim0[15:0]=0xFFFF
    g1[2] = (int)(0xFFFFu << 16);                     // tensor_dim0 hi=0 | tensor_dim1[15:0]=0xFFFF
    g1[3] = (int)(0x00FFu | (tile_x << 16));          // tensor_dim1 hi=0xFF | tile_dim0
    g1[4] = (int)tile_y;                              // tile_dim1 | tile_dim2=0
    g1[5] = (int)stride_elems;                        // tensor_dim0_stride[31:0]
    g1[6] = 0;                                        // stride0 hi | stride1 lo
    g1[7] = 0;
    i32x4 z4 = {0, 0, 0, 0};
#if defined(__clang_major__) && (__clang_major__ >= 23)
    i32x8 z8 = {0, 0, 0, 0, 0, 0, 0, 0};
    __builtin_amdgcn_tensor_load_to_lds(g0, g1, z4, z4, z8, 0);
#else
    __builtin_amdgcn_tensor_load_to_lds(g0, g1, z4, z4, 0);
#endif
}

// ---------------- kernel 0: zero the stats accumulators ----------------
__global__ void k_zero(float* __restrict__ p, int n) {
    int t = blockIdx.x * blockDim.x + threadIdx.x;
    if (t < n) p[t] = 0.f;
}

// ---------------- kernel 1: QKV GEMM (WMMA f32 16x16x4) + channel stats ----------------
// Per block b: qkv_b[o,i] = sum_c w_qkv[o,c] * x[n,c,i,w],  o<128, i<128, c<64
__global__ __launch_bounds__(256) void k_qkv(const float* __restrict__ x,
                                             const float* __restrict__ w,
                                             float* __restrict__ ws) {
    __shared__ float sA[kC2 * kC];     // w_qkv 128x64 row-major (o,c)
    __shared__ float sB[kC * kH];      // xh    64x128 row-major (c,i)
    __shared__ float sStat[2 * kC2];
    const int b = blockIdx.x;
    const int n = b >> 7, wcol = b & 127;
    const int tid = threadIdx.x;

    // TDM fills LDS: contiguous 32KB weight panel + strided x panel-gather
    // (lane-strided loads would hit a different 64B line per lane; the DMA
    // engine does the transpose-gather instead). Wave 0 issues & waits
    // TENSORcnt; barrier ordering makes it visible to all waves.
    if ((tid >> 5) == 0) {
        tdm_load_2d(w, lds_off(sA), kC2 * kC, 1, kC2 * kC);
        tdm_load_2d(x + (size_t)n * kC * kH * kW + wcol,
                    lds_off(sB), 1, kC * kH, kW);
        __builtin_amdgcn_s_wait_tensorcnt(0);
    }
    for (int t = tid; t < 2 * kC2; t += 256) sStat[t] = 0.f;
    __syncthreads();

    const int wv = tid >> 5, lane = tid & 31;
    const int m = lane & 15;                 // M (A) / N (B,D) lane index
    const int koff = (lane >> 4) * 2;        // K pair select per lane half
    const int hi8  = (lane >> 4) << 3;       // D: M += 8 for upper half
    const int o0 = wv * 16;

    float psum[8], psq[8];
#pragma unroll
    for (int r = 0; r < 8; ++r) { psum[r] = 0.f; psq[r] = 0.f; }

    float* qkv = ws + OFF_QKV + (size_t)b * (kC2 * kH);
    for (int jt = 0; jt < 8; ++jt) {
        const int j0 = jt * 16;
        v8f acc = {};
#pragma unroll
        for (int kk = 0; kk < 16; ++kk) {
            const int k0 = kk * 4 + koff;
            v2f a = *(const v2f*)&sA[(o0 + m) * kC + k0];   // A[m][k0], A[m][k0+1]
            v2f bb;
            bb.x = sB[k0 * kH + j0 + m];                    // B[k0][n]
            bb.y = sB[(k0 + 1) * kH + j0 + m];              // B[k0+1][n]
            acc = __builtin_amdgcn_wmma_f32_16x16x4_f32(false, a, false, bb,
                                                        (short)0, acc, false, false);
        }
#pragma unroll
        for (int r = 0; r < 8; ++r) {
            const int oo = o0 + r + hi8;
            const float v = acc[r];
            qkv[oo * kH + j0 + m] = v;
            psum[r] += v; psq[r] += v * v;
        }
    }
    // reduce within each 16-lane half (same channel), then LDS atomics
#pragma unroll
    for (int r = 0; r < 8; ++r) {
        float s = psum[r], q = psq[r];
        for (int off = 8; off >= 1; off >>= 1) {
            s += __shfl_xor(s, off, 32);
            q += __shfl_xor(q, off, 32);
        }
        if (m == 0) {
            const int oo = o0 + r + hi8;
            atomicAdd(&sStat[oo], s);
            atomicAdd(&sStat[kC2 + oo], q);
        }
    }
    __syncthreads();
    float* gs = ws + OFF_SQKV;
    if (tid < 2 * kC2) atomicAdd(&gs[tid], sStat[tid]);
}

// ---------------- kernel 2: turn (sum, sumsq) into BN (scale, shift) ----------------
__global__ void k_coef(const float* __restrict__ stat, float* __restrict__ coef,
                       const float* __restrict__ gamma, const float* __restrict__ beta,
                       int nch, float inv_count) {
    int t = blockIdx.x * blockDim.x + threadIdx.x;
    if (t >= nch) return;
    float mean = stat[t] * inv_count;
    float var  = stat[nch + t] * inv_count - mean * mean;
    float inv  = rsqrtf(var + kEPS);
    float sc   = gamma[t] * inv;
    coef[t]       = sc;
    coef[nch + t] = beta[t] - mean * sc;
}

// ---------------- kernel 3: sim stats pass (qk/qr/kr values, 24-channel stats) ----------------
__global__ __launch_bounds__(256) void k_simstat(const float* __restrict__ qkv_raw,
                                                 const float* __restrict__ coef,
                                                 const float* __restrict__ rel,
                                                 float* __restrict__ gstat) {
    __shared__ float sQK[8 * kH];      // normalized q rows 0..3, k rows 4..7
    __shared__ float sRel[8 * 255];
    __shared__ float sPart[8 * 6];
    const int idx = blockIdx.x, b = idx >> 3, g = idx & 7;
    const int tid = threadIdx.x;
    const float* qkv = qkv_raw + (size_t)b * (kC2 * kH);

    if ((tid >> 5) == 0) {             // TDM: contiguous relative-table stage
        tdm_load_2d(rel, lds_off(sRel), 8 * 255, 1, 8 * 255);
        __builtin_amdgcn_s_wait_tensorcnt(0);
    }
    for (int t = tid; t < 8 * kH; t += 256) {
        int cc = t >> 7, i = t & 127;
        int o = g * 16 + cc;
        sQK[t] = qkv[o * kH + i] * coef[o] + coef[kC2 + o];
    }
    __syncthreads();

    float sqk = 0, qqk = 0, sqr = 0, qqr = 0, skr = 0, qkr = 0;
    for (int e = tid; e < kH * kH; e += 256) {
        int i = e >> 7, j = e & 127;
        float aqk = 0, aqr = 0, akr = 0;
#pragma unroll
        for (int c = 0; c < 4; ++c) {
            float qv = sQK[c * kH + i];
            float kv = sQK[(4 + c) * kH + j];
            aqk += qv * kv;
            aqr += qv * sRel[c * 255 + (i - j + 127)];
            akr += kv * sRel[(4 + c) * 255 + (j - i + 127)];
        }
        aqr *= kF_QR; akr *= kF_KR;
        sqk += aqk; qqk += aqk * aqk;
        sqr += aqr; qqr += aqr * aqr;
        skr += akr; qkr += akr * akr;
    }
    for (int off = 16; off >= 1; off >>= 1) {
        sqk += __shfl_xor(sqk, off, 32); qqk += __shfl_xor(qqk, off, 32);
        sqr += __shfl_xor(sqr, off, 32); qqr += __shfl_xor(qqr, off, 32);
        skr += __shfl_xor(skr, off, 32); qkr += __shfl_xor(qkr, off, 32);
    }
    const int wv = tid >> 5;
    if ((tid & 31) == 0) {
        sPart[wv * 6 + 0] = sqk; sPart[wv * 6 + 1] = qqk;
        sPart[wv * 6 + 2] = sqr; sPart[wv * 6 + 3] = qqr;
        sPart[wv * 6 + 4] = skr; sPart[wv * 6 + 5] = qkr;
    }
    __syncthreads();
    if (tid < 6) {
        float acc = 0;
        for (int w8 = 0; w8 < 8; ++w8) acc += sPart[w8 * 6 + tid];
        int which = tid >> 1, issq = tid & 1;
        atomicAdd(&gstat[issq * 24 + which * 8 + g], acc);
    }
}

// ---------------- kernel 4: fused attention (logits WMMA + softmax + sv WMMA + sve) ----------------
__global__ __launch_bounds__(256) void k_attn(const float* __restrict__ qkv_raw,
                                              const float* __restrict__ cqkv,
                                              const float* __restrict__ csim,
                                              const float* __restrict__ rel,
                                              float* __restrict__ out_raw,
                                              float* __restrict__ gstat_out) {
    extern __shared__ float smem[];
    float* sSim  = smem;                  // 128*128
    float* sQKV  = sSim + kH * kH;        // 16*128  (q 0-3, k 4-7, v 8-15 normalized)
    float* sRel  = sQKV + 16 * kH;        // 16*255
    float* sStat = sRel + 16 * 255;       // 32
    const int idx = blockIdx.x, b = idx >> 3, g = idx & 7;
    const int tid = threadIdx.x;
    const float* qkv = qkv_raw + (size_t)b * (kC2 * kH);

    if ((tid >> 5) == 0) {             // TDM: contiguous relative-table stage
        tdm_load_2d(rel, lds_off(sRel), 16 * 255, 1, 16 * 255);
        __builtin_amdgcn_s_wait_tensorcnt(0);
    }
    for (int t = tid; t < 16 * kH; t += 256) {
        int cc = t >> 7, o = g * 16 + cc;
        sQKV[t] = qkv[o * kH + (t & 127)] * cqkv[o] + cqkv[kC2 + o];
    }
    if (tid < 32) sStat[tid] = 0.f;
    __syncthreads();

    const int wv = tid >> 5, lane = tid & 31;
    const int mm = lane & 15;
    const int koff = (lane >> 4) * 2;
    const int hi8  = (lane >> 4) << 3;
    const int i0 = wv * 16;

    const float csQK = csim[g],      cbQK = csim[24 + g];
    const float csQR = csim[8 + g],  cbQR = csim[24 + 8 + g];
    const float csKR = csim[16 + g], cbKR = csim[24 + 16 + g];

    // ---- logits: one WMMA (K=4) per 16x16 tile + VALU relative terms ----
    for (int jt = 0; jt < 8; ++jt) {
        const int j0 = jt * 16;
        v2f a, bb;
        a.x  = sQKV[(koff + 0) * kH + i0 + mm];     // A[m][k] = q[k][i0+m]
        a.y  = sQKV[(koff + 1) * kH + i0 + mm];
        bb.x = sQKV[(4 + koff + 0) * kH + j0 + mm]; // B[k][n] = k[k][j0+n]
        bb.y = sQKV[(4 + koff + 1) * kH + j0 + mm];
        v8f acc = {};
        acc = __builtin_amdgcn_wmma_f32_16x16x4_f32(false, a, false, bb,
                                                    (short)0, acc, false, false);
#pragma unroll
        for (int r = 0; r < 8; ++r) {
            const int i = i0 + r + hi8, j = j0 + mm;
            float aqr = 0.f, akr = 0.f;
#pragma unroll
            for (int c = 0; c < 4; ++c) {
                aqr += sQKV[c * kH + i] * sRel[c * 255 + (i - j + 127)];
                akr += sQKV[(4 + c) * kH + j] * sRel[(4 + c) * 255 + (j - i + 127)];
            }
            sSim[i * kH + j] = csQK * acc[r] + cbQK
                             + csQR * (aqr * kF_QR) + cbQR
                             + csKR * (akr * kF_KR) + cbKR;
        }
    }
    __syncthreads();

    // ---- softmax over j, wave-local rows ----
    for (int rr = 0; rr < 16; ++rr) {
        const int i = i0 + rr;
        float v0 = sSim[i * kH + lane];
        float v1 = sSim[i * kH + lane + 32];
        float v2 = sSim[i * kH + lane + 64];
        float v3 = sSim[i * kH + lane + 96];
        float mx = fmaxf(fmaxf(v0, v1), fmaxf(v2, v3));
        for (int off = 16; off >= 1; off >>= 1) mx = fmaxf(mx, __shfl_xor(mx, off, 32));
        v0 = __builtin_amdgcn_exp2f((v0 - mx) * kLOG2E);
        v1 = __builtin_amdgcn_exp2f((v1 - mx) * kLOG2E);
        v2 = __builtin_amdgcn_exp2f((v2 - mx) * kLOG2E);
        v3 = __builtin_amdgcn_exp2f((v3 - mx) * kLOG2E);
        float sm = v0 + v1 + v2 + v3;
        for (int off = 16; off >= 1; off >>= 1) sm += __shfl_xor(sm, off, 32);
        const float inv = __builtin_amdgcn_rcpf(sm);
        sSim[i * kH + lane]      = v0 * inv;
        sSim[i * kH + lane + 32] = v1 * inv;
        sSim[i * kH + lane + 64] = v2 * inv;
        sSim[i * kH + lane + 96] = v3 * inv;
    }
    __syncthreads();

    float* orow = out_raw + (size_t)b * (kC2 * kH);

    // ---- sv = sim @ v^T via WMMA, K=128 (N=c padded 8->16) ----
    {
        v8f acc = {};
#pragma unroll 4
        for (int kt = 0; kt < 32; ++kt) {
            const int k0 = kt * 4 + koff;
            v2f a = *(const v2f*)&sSim[(i0 + mm) * kH + k0];
            v2f bb;
            if (mm < 8) bb = *(const v2f*)&sQKV[(8 + mm) * kH + k0];
            else { bb.x = 0.f; bb.y = 0.f; }
            acc = __builtin_amdgcn_wmma_f32_16x16x4_f32(false, a, false, bb,
                                                        (short)0, acc, false, false);
        }
        float psum = 0.f, psq = 0.f;
#pragma unroll
        for (int r = 0; r < 8; ++r) {
            const float v = acc[r] * kF_SV;
            const int i = i0 + r + hi8;
            if (mm < 8) orow[(g * 8 + mm) * kH + i] = v;
            psum += v; psq += v * v;
        }
        psum += __shfl_xor(psum, 16, 32);
        psq  += __shfl_xor(psq, 16, 32);
        if (lane < 8) {
            atomicAdd(&sStat[lane], psum);
            atomicAdd(&sStat[16 + lane], psq);
        }
    }

    // ---- sve[c,i] = F_SVE * sum_j sim[i,j] * v_emb[c, i-j+127] (VALU) ----
#pragma unroll
    for (int tt = 0; tt < 4; ++tt) {
        const int task = lane + 32 * tt;     // 128 tasks per wave: (c, i-in-tile)
        const int c = task >> 4;
        const int i = i0 + (task & 15);
        float acc = 0.f;
        for (int j = 0; j < kH; ++j)
            acc += sSim[i * kH + j] * sRel[(8 + c) * 255 + (i - j + 127)];
        const float v = acc * kF_SVE;
        orow[(64 + g * 8 + c) * kH + i] = v;
        atomicAdd(&sStat[8 + c], v);
        atomicAdd(&sStat[24 + c], v * v);
    }
    __syncthreads();
    if (tid < 32) {
        const int lc = tid & 15;
        const int o = (lc < 8) ? (g * 8 + lc) : (64 + g * 8 + (lc - 8));
        atomicAdd(&gstat_out[((tid < 16) ? 0 : kC2) + o], sStat[tid]);
    }
}

// ---------------- kernel 5: final BN + pair-sum + transpose to NCHW ----------------
__global__ __launch_bounds__(256) void k_final(const float* __restrict__ out_raw,
                                               const float* __restrict__ co,
                                               float* __restrict__ out) {
    __shared__ float tile[16][17];
    int idx = blockIdx.x;
    const int wb = idx & 7; idx >>= 3;
    const int hb = idx & 7; idx >>= 3;
    const int m  = idx & 63; idx >>= 6;
    const int n  = idx;
    const int tid = threadIdx.x;
    const int ty = tid >> 4, tx = tid & 15;
    {
        const int w = wb * 16 + ty, h = hb * 16 + tx;
        const float* orow = out_raw + (size_t)(n * 128 + w) * (kC2 * kH);
        const int o0 = 2 * m;
        const float r0 = orow[o0 * kH + h];
        const float r1 = orow[(o0 + 1) * kH + h];
        tile[ty][tx] = co[o0] * r0 + co[kC2 + o0] + co[o0 + 1] * r1 + co[kC2 + o0 + 1];
    }
    __syncthreads();
    {
        const int h = hb * 16 + ty, w = wb * 16 + tx;
        out[(((size_t)n * kC + m) * kH + h) * kW + w] = tile[tx][ty];
    }
}

// ---------------- host-side launch ----------------
extern "C" void kernel_launch(void* const* d_in, const int* in_sizes, int n_in,
                              void* d_out, int out_size, void* d_ws, size_t ws_size,
                              hipStream_t stream) {
    const float* x     = (const float*)d_in[0];
    const float* w_qkv = (const float*)d_in[1];
    const float* g_qkv = (const float*)d_in[2];
    const float* b_qkv = (const float*)d_in[3];
    const float* g_sim = (const float*)d_in[4];
    const float* b_sim = (const float*)d_in[5];
    const float* g_out = (const float*)d_in[6];
    const float* b_out = (const float*)d_in[7];
    const float* rel   = (const float*)d_in[8];
    float* ws  = (float*)d_ws;
    float* out = (float*)d_out;

    // zero all stats accumulators (560 floats starting at OFF_SQKV)
    k_zero<<<3, 256, 0, stream>>>(ws + OFF_SQKV, 2 * kC2 + 48 + 2 * kC2);

    k_qkv<<<kB, 256, 0, stream>>>(x, w_qkv, ws);
    k_coef<<<1, 128, 0, stream>>>(ws + OFF_SQKV, ws + OFF_CQKV, g_qkv, b_qkv,
                                  kC2, 1.f / (kB * kH));
    k_simstat<<<kB * kG, 256, 0, stream>>>(ws + OFF_QKV, ws + OFF_CQKV, rel,
                                           ws + OFF_SSIM);
    k_coef<<<1, 64, 0, stream>>>(ws + OFF_SSIM, ws + OFF_CSIM, g_sim, b_sim,
                                 24, 1.f / ((float)kB * kH * kH));
    const size_t smem = (size_t)(kH * kH + 16 * kH + 16 * 255 + 32) * sizeof(float);
    k_attn<<<kB * kG, 256, smem, stream>>>(ws + OFF_QKV, ws + OFF_CQKV,
                                           ws + OFF_CSIM, rel,
                                           ws + OFF_OUTR, ws + OFF_SOUT);
    k_coef<<<1, 128, 0, stream>>>(ws + OFF_SOUT, ws + OFF_COUT, g_out, b_out,
                                  kC2, 1.f / (kB * kH));
    k_final<<<kN * kC * 8 * 8, 256, 0, stream>>>(ws + OFF_OUTR, ws + OFF_COUT, out);
}